// NLL_10316511445616
// MI455X (gfx1250) — compile-verified
//
#include <hip/hip_runtime.h>

// NLL loss: out = -mean_i( log( x[i, y[i]] ) ),  x: [N,C] f32, y: [N] labels.
// Optimal MI455X strategy: this is a latency-bound gather (32 KB touched out
// of 1 GB). One scattered load per row, wave32 shuffle tree reduce, two-stage
// deterministic reduction (no float atomics -> bitwise-stable across replays).

#define BLOCK 128            // 4 wave32 waves per block
#define MAXWAVES (BLOCK / 32)

__device__ __forceinline__ float wave_reduce_add(float v) {
    // CDNA5 is wave32: tree starts at offset 16 (offset 32 would be a silent
    // CDNA4 wave64 bug).
    #pragma unroll
    for (int off = 16; off > 0; off >>= 1)
        v += __shfl_down(v, off, 32);
    return v;
}

__global__ void nll_gather_partial(const float* __restrict__ x,
                                   const int*  __restrict__ y,
                                   float* __restrict__ partial,
                                   int n, int c) {
    const int i = blockIdx.x * BLOCK + threadIdx.x;

    float v = 0.0f;
    if (i < n) {
        const int cls = y[i];                        // class id, fits int32
        // 64-bit byte offset: N*C*4 ~ 1.05 GB still fits signed 32-bit; size_t guards larger shapes.
        const float p = x[(size_t)i * (size_t)c + (size_t)cls];
        v = -__logf(p) ;                             // accumulate -log(p)
        v = -logf(p);                                // precise log (match ref)
    }

    // Stage 1: per-wave reduce (deterministic fixed tree order).
    v = wave_reduce_add(v);

    // Stage 2: cross-wave reduce through LDS (ds_store/ds_load + barrier).
    __shared__ float lds[MAXWAVES];
    const int lane = threadIdx.x & 31;
    const int wave = threadIdx.x >> 5;
    if (lane == 0) lds[wave] = v;
    __syncthreads();

    if (wave == 0) {
        float s = (lane < MAXWAVES) ? lds[lane] : 0.0f;
        s = wave_reduce_add(s);
        if (lane == 0) partial[blockIdx.x] = s;      // one partial per block
    }
}

__global__ void nll_finalize(const float* __restrict__ partial,
                             float* __restrict__ out,
                             int nblocks, float inv_n) {
    // Single wave32; fixed-order fold over block partials -> deterministic.
    const int lane = threadIdx.x;
    float s = 0.0f;
    for (int j = lane; j < nblocks; j += 32)
        s += partial[j];
    s = wave_reduce_add(s);
    if (lane == 0)
        out[0] = s * inv_n;                          // mean of -log(p)
}

extern "C" void kernel_launch(void* const* d_in, const int* in_sizes, int n_in,
                              void* d_out, int out_size, void* d_ws, size_t ws_size,
                              hipStream_t stream) {
    const float* x = (const float*)d_in[0];   // [N*C] f32 probabilities
    const int*   y = (const int*)d_in[1];     // [N] labels (harness int cast)
    float* out = (float*)d_out;               // scalar f32

    const int n = in_sizes[1];                // N = 8192
    const int c = in_sizes[0] / in_sizes[1];  // C = 32000

    const int nblocks = (n + BLOCK - 1) / BLOCK;   // 64 blocks x 128 thr = 8192
    float* partial = (float*)d_ws;                 // 64 floats of scratch

    nll_gather_partial<<<nblocks, BLOCK, 0, stream>>>(x, y, partial, n, c);
    nll_finalize<<<1, 32, 0, stream>>>(partial, out, nblocks, 1.0f / (float)n);
}